// ECELoss_25804163514418
// MI455X (gfx1250) — compile-verified
//
#include <hip/hip_runtime.h>
#include <math.h>

#define N_ROWS      262144
#define N_CLASSES   1000
#define N_F4        250            // 1000 floats = 250 float4
#define N_BINS      15
#define BLOCKS      1024
#define WPB         8              // waves per block (256 threads, wave32)
#define TOTAL_WAVES (BLOCKS * WPB) // 8192
#define ROWS_PER_WAVE (N_ROWS / TOTAL_WAVES) // 32
#define LDS_ROW_F   1008           // padded row (floats), 16B-multiple stride

// GVS-mode async global->LDS B128: global = SADDR + VADDR + offset,
// LDS dest = VDST + offset (INST_OFFSET applies to both sides).
#define ASYNC_B128(ldsA, voff, sbase, imm)                                  \
    asm volatile("global_load_async_to_lds_b128 %0, %1, %2 offset:%3"       \
                 :: "v"(ldsA), "v"(voff), "s"(sbase), "i"(imm) : "memory")

// Issue 8 async B128 transfers covering one row (1000 floats = 250 float4).
// t=0..6 share one per-lane VADDR (lane*16) with immediate offsets t*512;
// t=7 uses a clamped VADDR (tail lanes duplicate float4 #249 -> benign).
__device__ __forceinline__ void async_load_row(unsigned long long sbase,
                                               unsigned ldsA, unsigned ldsA7,
                                               unsigned voff, unsigned voff7) {
    ASYNC_B128(ldsA,  voff,  sbase, 0 * 512);
    ASYNC_B128(ldsA,  voff,  sbase, 1 * 512);
    ASYNC_B128(ldsA,  voff,  sbase, 2 * 512);
    ASYNC_B128(ldsA,  voff,  sbase, 3 * 512);
    ASYNC_B128(ldsA,  voff,  sbase, 4 * 512);
    ASYNC_B128(ldsA,  voff,  sbase, 5 * 512);
    ASYNC_B128(ldsA,  voff,  sbase, 6 * 512);
    ASYNC_B128(ldsA7, voff7, sbase, 0);
}

__global__ __launch_bounds__(256) void ece_partial_kernel(
        const float* __restrict__ logits,
        const int*   __restrict__ labels,
        float*       __restrict__ ws) {
    __shared__ __align__(16) float lds[WPB][2][LDS_ROW_F];

    const int lane = threadIdx.x & 31;
    const int wid  = threadIdx.x >> 5;
    const int gw   = blockIdx.x * WPB + wid;   // global wave id

    // Per-lane address components, invariant across rows.
    int e7 = 224 + lane; if (e7 > N_F4 - 1) e7 = N_F4 - 1;   // tail clamp
    const unsigned voff  = (unsigned)lane * 16u;             // t=0..6 VADDR
    const unsigned voff7 = (unsigned)e7 * 16u;               // t=7   VADDR

    unsigned ldsA[2], ldsA7[2];
    ldsA[0]  = (unsigned)(size_t)(&lds[wid][0][0]) + voff;
    ldsA[1]  = (unsigned)(size_t)(&lds[wid][1][0]) + voff;
    ldsA7[0] = (unsigned)(size_t)(&lds[wid][0][0]) + voff7;
    ldsA7[1] = (unsigned)(size_t)(&lds[wid][1][0]) + voff7;

    float binAcc = 0.0f;   // lane l accumulates signed sum for bin l (l<15)

    // Prefetch first row into buffer 0.
    async_load_row((unsigned long long)(logits + (size_t)gw * N_CLASSES),
                   ldsA[0], ldsA7[0], voff, voff7);

    for (int k = 0; k < ROWS_PER_WAVE; ++k) {
        const int r   = gw + k * TOTAL_WAVES;
        const int buf = k & 1;

        if (k + 1 < ROWS_PER_WAVE) {
            const int rn = gw + (k + 1) * TOTAL_WAVES;
            async_load_row((unsigned long long)(logits + (size_t)rn * N_CLASSES),
                           ldsA[buf ^ 1], ldsA7[buf ^ 1], voff, voff7);
            // 16 outstanding, in-order completion: <=8 means current row done.
            asm volatile("s_wait_asynccnt 0x8" ::: "memory");
        } else {
            asm volatile("s_wait_asynccnt 0x0" ::: "memory");
        }

        const float* row = &lds[wid][buf][0];

        // Pass 1 over LDS: max + first-occurrence argmax.
        float m = -INFINITY;
        int   am = 0x7fffffff;
        for (int j = lane; j < N_CLASSES; j += 32) {
            float v = row[j];
            if (v > m) { m = v; am = j; }
        }
#pragma unroll
        for (int off = 16; off >= 1; off >>= 1) {
            float om = __shfl_xor(m, off, 32);
            int   oa = __shfl_xor(am, off, 32);
            if (om > m || (om == m && oa < am)) { m = om; am = oa; }
        }

        // Pass 2 over LDS: sum of exp(x - m).
        float s = 0.0f;
        for (int j = lane; j < N_CLASSES; j += 32) s += __expf(row[j] - m);
#pragma unroll
        for (int off = 16; off >= 1; off >>= 1) s += __shfl_xor(s, off, 32);

        float conf = 1.0f / s;                       // max softmax prob
        float acc  = (am == labels[r]) ? 1.0f : 0.0f;
        int bin = (int)ceilf(conf * (float)N_BINS) - 1;
        bin = bin < 0 ? 0 : (bin > N_BINS - 1 ? N_BINS - 1 : bin);
        float diff = conf - acc;
        if (lane == bin) binAcc += diff;
    }

    if (lane < N_BINS)
        ws[(size_t)gw * N_BINS + lane] = binAcc;
}

__global__ __launch_bounds__(256) void ece_final_kernel(
        const float* __restrict__ ws, float* __restrict__ out) {
    __shared__ float red[256];
    const int t = threadIdx.x;
    float total = 0.0f;
    for (int b = 0; b < N_BINS; ++b) {
        float p = 0.0f;
        for (int w = t; w < TOTAL_WAVES; w += 256)
            p += ws[(size_t)w * N_BINS + b];
        red[t] = p;
        __syncthreads();
        for (int off = 128; off >= 1; off >>= 1) {
            if (t < off) red[t] += red[t + off];
            __syncthreads();
        }
        if (t == 0) total += fabsf(red[0]);
        __syncthreads();
    }
    if (t == 0) out[0] = total / (float)N_ROWS;
}

extern "C" void kernel_launch(void* const* d_in, const int* in_sizes, int n_in,
                              void* d_out, int out_size, void* d_ws, size_t ws_size,
                              hipStream_t stream) {
    const float* logits = (const float*)d_in[0];
    const int*   labels = (const int*)d_in[1];
    float* ws  = (float*)d_ws;          // needs 8192*15*4 = 480 KB
    float* out = (float*)d_out;

    ece_partial_kernel<<<BLOCKS, 256, 0, stream>>>(logits, labels, ws);
    ece_final_kernel<<<1, 256, 0, stream>>>(ws, out);
}